// LIFSpike_70720931496905
// MI455X (gfx1250) — compile-verified
//
#include <hip/hip_runtime.h>

// LIF spike recurrence, memory-bound streaming kernel for MI455X (gfx1250).
// x: [T=8, B=32, C=128, H=32, W=32] fp32 -> spikes same shape fp32.
// mem_{t} = mem_{t-1}*0.5 + x_t*alpha ; spike = mem > Vth ; mem = spike?0:mem

typedef float f4 __attribute__((ext_vector_type(4)));

#define LIF_T 8
#define LIF_TAU 0.5f

__global__ __launch_bounds__(256) void lif_spike_kernel(
    const f4* __restrict__ x,
    const float* __restrict__ alpha_p,
    const float* __restrict__ vth_p,
    f4* __restrict__ out,
    unsigned n4)  // number of f4 vectors per timestep (spatial/4)
{
    unsigned i = blockIdx.x * blockDim.x + threadIdx.x;
    if (i >= n4) return;

    const float alpha = alpha_p[0];   // uniform -> s_load_b32
    const float vth   = vth_p[0];     // uniform -> s_load_b32

    // Issue all 8 timestep loads before the dependent chain: 8x
    // global_load_b128 th:NT in flight per thread (max MLP, one waitcnt ladder).
    f4 xr[LIF_T];
#pragma unroll
    for (int t = 0; t < LIF_T; ++t) {
        xr[t] = __builtin_nontemporal_load(&x[(size_t)t * n4 + i]);
    }

    f4 mem;
    mem.x = 0.0f; mem.y = 0.0f; mem.z = 0.0f; mem.w = 0.0f;

#pragma unroll
    for (int t = 0; t < LIF_T; ++t) {
        f4 m;
        m.x = mem.x * LIF_TAU + xr[t].x * alpha;
        m.y = mem.y * LIF_TAU + xr[t].y * alpha;
        m.z = mem.z * LIF_TAU + xr[t].z * alpha;
        m.w = mem.w * LIF_TAU + xr[t].w * alpha;

        f4 sp;
        sp.x = (m.x > vth) ? 1.0f : 0.0f;   // v_cmp + v_cndmask
        sp.y = (m.y > vth) ? 1.0f : 0.0f;
        sp.z = (m.z > vth) ? 1.0f : 0.0f;
        sp.w = (m.w > vth) ? 1.0f : 0.0f;

        // hard reset on spike
        mem.x = (m.x > vth) ? 0.0f : m.x;
        mem.y = (m.y > vth) ? 0.0f : m.y;
        mem.z = (m.z > vth) ? 0.0f : m.z;
        mem.w = (m.w > vth) ? 0.0f : m.w;

        __builtin_nontemporal_store(sp, &out[(size_t)t * n4 + i]);  // b128 th:NT
    }
}

extern "C" void kernel_launch(void* const* d_in, const int* in_sizes, int n_in,
                              void* d_out, int out_size, void* d_ws, size_t ws_size,
                              hipStream_t stream) {
    const float* x     = (const float*)d_in[0];  // [T,B,C,H,W] fp32
    const float* alpha = (const float*)d_in[1];  // scalar
    const float* vth   = (const float*)d_in[2];  // scalar
    float* out         = (float*)d_out;

    const unsigned total   = (unsigned)in_sizes[0];      // 33,554,432
    const unsigned spatial = total / LIF_T;              // 4,194,304 (B*C*H*W)
    const unsigned n4      = spatial / 4;                // 1,048,576 f4 vectors

    const unsigned block = 256;
    const unsigned grid  = (n4 + block - 1) / block;     // 4096 blocks

    lif_spike_kernel<<<grid, block, 0, stream>>>(
        (const f4*)x, alpha, vth, (f4*)out, n4);
}